// AtomBondConsistencyLayer_35184372089134
// MI455X (gfx1250) — compile-verified
//
#include <hip/hip_runtime.h>
#include <math.h>

#define NN 1024
#define DD 256
#define ATYPES 119
#define BTYPES 5
#define NPAIRS ((NN * (NN - 1)) / 2)   // 523776 = 16 * 32736 exactly

typedef __attribute__((ext_vector_type(2))) float v2f;
typedef __attribute__((ext_vector_type(8))) float v8f;

__device__ __forceinline__ v8f wmma_f32_16x16x4(v2f a, v2f b, v8f c) {
  // D = A(16x4,f32) * B(4x16,f32) + C(16x16,f32); full-precision matrix op
  return __builtin_amdgcn_wmma_f32_16x16x4_f32(false, a, false, b, (short)0, c,
                                               false, false);
}

__device__ __forceinline__ float silu_f(float x) {
  return x * (1.0f / (1.0f + __expf(-x)));
}

// ---------------------------------------------------------------------------
// Generic wave-per-16x16-tile GEMM: C = act(A[MxK] @ W[KxN] + bias)
// A row-major (lda), W row-major (ldw), C row-major (ldc). K % 4 == 0.
// A-frag: lanes 0-15 row M=lane, VGPR0=K, VGPR1=K+1; lanes 16-31 K+2,K+3.
// B-frag: lanes 0-15 col N=lane rows K,K+1; lanes 16-31 rows K+2,K+3.
// Out-of-range columns: B column n only feeds D column n, and we never store
// those D columns, so we just CLAMP the load index (no masking, no branches).
// ---------------------------------------------------------------------------
__global__ void gemm16_wmma(const float* __restrict__ A,
                            const float* __restrict__ W,
                            const float* __restrict__ bias,
                            float* __restrict__ C,
                            int M, int K, int Ncols,
                            int lda, int ldw, int ldc,
                            int colTiles, int act) {
  int gw = (blockIdx.x * blockDim.x + threadIdx.x) >> 5;
  int lane = threadIdx.x & 31;
  int rowTile = gw / colTiles;
  int colTile = gw - rowTile * colTiles;
  int row0 = rowTile << 4;
  int col0 = colTile << 4;
  if (row0 >= M) return;

  int m = lane & 15;
  int half = lane >> 4;
  int koff = half << 1;
  int c = col0 + m;
  bool cOK = c < Ncols;
  int cc = cOK ? c : (Ncols - 1);  // clamped, always in-bounds

  const float* arow = A + (size_t)(row0 + m) * lda + koff;
  const float* wp0 = W + (size_t)koff * ldw + cc;        // row k+koff
  const float* wp1 = wp0 + ldw;                          // row k+koff+1
  size_t wstep = (size_t)ldw * 4;

  v8f acc = {};
  for (int k = 0; k < K; k += 4) {
    v2f a = *reinterpret_cast<const v2f*>(arow + k);
    v2f b;
    b.x = *wp0;
    b.y = *wp1;
    wp0 += wstep;
    wp1 += wstep;
    acc = wmma_f32_16x16x4(a, b, acc);
  }
  float bv = (bias != nullptr) ? bias[cc] : 0.0f;
#pragma unroll
  for (int r = 0; r < 8; ++r) {
    int row = row0 + r + (half << 3);
    float v = acc[r] + bv;
    if (act == 1) v = silu_f(v);
    if (cOK) C[(size_t)row * ldc + c] = v;
  }
}

// ---------------------------------------------------------------------------
// Row softmax over 119 logits, one wave per row (wave32).
// ---------------------------------------------------------------------------
__global__ void softmax119(const float* __restrict__ logits,
                           float* __restrict__ probs) {
  int gw = (blockIdx.x * blockDim.x + threadIdx.x) >> 5;
  int lane = threadIdx.x & 31;
  if (gw >= NN) return;
  const float* row = logits + (size_t)gw * ATYPES;
  float v[4];
  float mx = -3.4e38f;
#pragma unroll
  for (int t = 0; t < 4; ++t) {
    int cix = lane + t * 32;
    v[t] = (cix < ATYPES) ? row[cix] : -3.4e38f;
    mx = fmaxf(mx, v[t]);
  }
#pragma unroll
  for (int s = 16; s >= 1; s >>= 1) mx = fmaxf(mx, __shfl_xor(mx, s, 32));
  float sum = 0.0f;
#pragma unroll
  for (int t = 0; t < 4; ++t) {
    int cix = lane + t * 32;
    v[t] = (cix < ATYPES) ? __expf(v[t] - mx) : 0.0f;
    sum += v[t];
  }
#pragma unroll
  for (int s = 16; s >= 1; s >>= 1) sum += __shfl_xor(sum, s, 32);
  float inv = 1.0f / sum;
  float* prow = probs + (size_t)gw * ATYPES;
#pragma unroll
  for (int t = 0; t < 4; ++t) {
    int cix = lane + t * 32;
    if (cix < ATYPES) prow[cix] = v[t] * inv;
  }
}

// ---------------------------------------------------------------------------
// valency_hidden = silu(h @ Wv1[:256] + probs @ Wv1[256:375] + bv1)  [1024x256]
// K padded 119 -> 120: only the A side must be zeroed (A=0 kills any B value),
// so A uses clamped-address load + value select; B loads are unconditional.
// ---------------------------------------------------------------------------
__global__ void valency_hidden_wmma(const float* __restrict__ h,
                                    const float* __restrict__ probs,
                                    const float* __restrict__ Wv1,
                                    const float* __restrict__ bv1,
                                    float* __restrict__ V1) {
  const int colTiles = DD / 16;
  int gw = (blockIdx.x * blockDim.x + threadIdx.x) >> 5;
  int lane = threadIdx.x & 31;
  int rowTile = gw / colTiles;
  int col0 = (gw - rowTile * colTiles) << 4;
  int row0 = rowTile << 4;
  if (row0 >= NN) return;

  int m = lane & 15;
  int half = lane >> 4;
  int koff = half << 1;
  int c = col0 + m;
  const float* hrow = h + (size_t)(row0 + m) * DD + koff;
  const float* prow = probs + (size_t)(row0 + m) * ATYPES;

  v8f acc = {};
  {
    const float* wp0 = Wv1 + (size_t)koff * DD + c;
    const float* wp1 = wp0 + DD;
    for (int k = 0; k < DD; k += 4) {
      v2f a = *reinterpret_cast<const v2f*>(hrow + k);
      v2f b;
      b.x = *wp0;
      b.y = *wp1;
      wp0 += (size_t)DD * 4;
      wp1 += (size_t)DD * 4;
      acc = wmma_f32_16x16x4(a, b, acc);
    }
  }
  {
    const float* Wv1b = Wv1 + (size_t)DD * DD;  // rows 256..374
    for (int k = 0; k < 120; k += 4) {          // 119 padded to 120
      int k0 = k + koff, k1 = k0 + 1;
      int kc0 = (k0 < ATYPES) ? k0 : (ATYPES - 1);
      int kc1 = (k1 < ATYPES) ? k1 : (ATYPES - 1);
      v2f a, b;
      a.x = (k0 < ATYPES) ? prow[kc0] : 0.0f;   // value select -> v_cndmask
      a.y = (k1 < ATYPES) ? prow[kc1] : 0.0f;
      b.x = Wv1b[(size_t)kc0 * DD + c];
      b.y = Wv1b[(size_t)kc1 * DD + c];
      acc = wmma_f32_16x16x4(a, b, acc);
    }
  }
  float bv = bv1[c];
#pragma unroll
  for (int r = 0; r < 8; ++r) {
    int row = row0 + r + (half << 3);
    V1[(size_t)row * DD + c] = silu_f(acc[r] + bv);
  }
}

// ---------------------------------------------------------------------------
// valency_scores = V1 @ Wv2 + bv2   [1024,1]; one thread per row (tiny)
// ---------------------------------------------------------------------------
__global__ void valency_out(const float* __restrict__ V1,
                            const float* __restrict__ Wv2,
                            const float* __restrict__ bv2,
                            float* __restrict__ out) {
  int row = blockIdx.x * blockDim.x + threadIdx.x;
  if (row >= NN) return;
  const float* vr = V1 + (size_t)row * DD;
  float s = 0.0f;
  for (int k = 0; k < DD; k += 4)
    s += vr[k] * Wv2[k] + vr[k + 1] * Wv2[k + 1] + vr[k + 2] * Wv2[k + 2] +
         vr[k + 3] * Wv2[k + 3];
  out[row] = s + bv2[0];
}

// ---------------------------------------------------------------------------
// Bond kernel: each wave handles 16 consecutive pairs p0..p0+15.
// bond_logits[p] = silu(Apre[i] + Bpre[j]) @ Wb2 + bb2, fused into WMMA A-frag.
// Wb2 (256x5, zero-padded to x16) staged in LDS as interleaved K-pairs so the
// B-fragment is a single 8-byte ds load per K-step.
// ---------------------------------------------------------------------------
__global__ void bond_wmma(const float* __restrict__ Apre,
                          const float* __restrict__ Bpre,
                          const float* __restrict__ Wb2,
                          const float* __restrict__ bb2,
                          float* __restrict__ bond_out,
                          float* __restrict__ pair_out) {
  __shared__ v2f sB[(DD / 2) * 16];  // sB[kp*16+c] = {Wb2p[2kp][c], Wb2p[2kp+1][c]}
  int tid = threadIdx.x;
  for (int e = tid; e < (DD / 2) * 16; e += blockDim.x) {
    int kp = e >> 4;
    int cc = e & 15;
    v2f w;
    w.x = (cc < BTYPES) ? Wb2[(size_t)(2 * kp) * BTYPES + cc] : 0.0f;
    w.y = (cc < BTYPES) ? Wb2[(size_t)(2 * kp + 1) * BTYPES + cc] : 0.0f;
    sB[e] = w;
  }
  __syncthreads();

  int wave = tid >> 5;
  int lane = tid & 31;
  int tile = blockIdx.x * (blockDim.x >> 5) + wave;
  int p0 = tile << 4;
  if (p0 >= NPAIRS) return;

  int m = lane & 15;
  int half = lane >> 4;
  int koff = half << 1;
  int pm = p0 + m;

  // decode triangular pair index: p = i*(2N-1-i)/2 + (j-i-1)
  int i, j;
  {
    double bq = 2.0 * (double)NN - 1.0;
    int ii = (int)((bq - sqrt(bq * bq - 8.0 * (double)pm)) * 0.5);
    if (ii < 0) ii = 0;
    if (ii > NN - 2) ii = NN - 2;
    while ((ii + 1) * (2 * NN - 1 - (ii + 1)) / 2 <= pm) ++ii;
    while (ii * (2 * NN - 1 - ii) / 2 > pm) --ii;
    int cum = ii * (2 * NN - 1 - ii) / 2;
    i = ii;
    j = ii + 1 + (pm - cum);
  }
  const float* ra = Apre + (size_t)i * DD + koff;  // bb1 folded into Apre
  const float* rb = Bpre + (size_t)j * DD + koff;

  // gfx1250 global_prefetch_b8: pull both 1KB rows (4 cachelines each)
#pragma unroll
  for (int pl = 0; pl < 4; ++pl) {
    __builtin_prefetch(ra - koff + pl * 64, 0, 3);
    __builtin_prefetch(rb - koff + pl * 64, 0, 3);
  }

  v8f acc = {};
  for (int k = 0; k < DD; k += 4) {
    v2f xa = *reinterpret_cast<const v2f*>(ra + k);
    v2f xb = *reinterpret_cast<const v2f*>(rb + k);
    v2f a;
    a.x = silu_f(xa.x + xb.x);
    a.y = silu_f(xa.y + xb.y);
    v2f b = sB[((k + koff) >> 1) * 16 + m];
    acc = wmma_f32_16x16x4(a, b, acc);
  }

  float bias = (m < BTYPES) ? bb2[m] : 0.0f;
#pragma unroll
  for (int r = 0; r < 8; ++r) {
    int p = p0 + r + (half << 3);
    if (m < BTYPES) bond_out[(size_t)p * BTYPES + m] = acc[r] + bias;
  }
  if (half == 0) {  // atom_pairs written as float values into the float d_out
    pair_out[(size_t)pm * 2 + 0] = (float)i;
    pair_out[(size_t)pm * 2 + 1] = (float)j;
  }
}

// ---------------------------------------------------------------------------
extern "C" void kernel_launch(void* const* d_in, const int* in_sizes, int n_in,
                              void* d_out, int out_size, void* d_ws,
                              size_t ws_size, hipStream_t stream) {
  (void)in_sizes; (void)n_in; (void)out_size; (void)ws_size;
  const float* h   = (const float*)d_in[0];
  // d_in[1] = edge_index (int64), d_in[2] = batch (int64): unused by reference
  const float* Wa1 = (const float*)d_in[3];   // [256,128]
  const float* ba1 = (const float*)d_in[4];   // [128]
  const float* Wa2 = (const float*)d_in[5];   // [128,119]
  const float* ba2 = (const float*)d_in[6];   // [119]
  const float* Wb1 = (const float*)d_in[7];   // [512,256]
  const float* bb1 = (const float*)d_in[8];   // [256]
  const float* Wb2 = (const float*)d_in[9];   // [256,5]
  const float* bb2 = (const float*)d_in[10];  // [5]
  const float* Wv1 = (const float*)d_in[11];  // [375,256]
  const float* bv1 = (const float*)d_in[12];  // [256]
  const float* Wv2 = (const float*)d_in[13];  // [256,1]
  const float* bv2 = (const float*)d_in[14];  // [1]

  float* out = (float*)d_out;
  float* atom_logits = out;                               // 1024*119
  float* bond_logits = out + (size_t)NN * ATYPES;         // P*5
  float* valency = bond_logits + (size_t)NPAIRS * BTYPES; // 1024
  float* pairs = valency + NN;                            // P*2

  float* ws = (float*)d_ws;
  float* H1    = ws;                          // 1024*128
  float* probs = H1 + (size_t)NN * 128;       // 1024*119
  float* V1    = probs + (size_t)NN * ATYPES; // 1024*256
  float* Apre  = V1 + (size_t)NN * DD;        // 1024*256
  float* Bpre  = Apre + (size_t)NN * DD;      // 1024*256

  // atom MLP
  gemm16_wmma<<<64, 256, 0, stream>>>(h, Wa1, ba1, H1, NN, DD, 128, DD, 128,
                                      128, 8, 1);
  gemm16_wmma<<<64, 256, 0, stream>>>(H1, Wa2, ba2, atom_logits, NN, 128,
                                      ATYPES, 128, ATYPES, ATYPES, 8, 0);
  softmax119<<<128, 256, 0, stream>>>(atom_logits, probs);

  // bond precompute: Apre = h@Wb1_top + bb1 ; Bpre = h@Wb1_bot
  gemm16_wmma<<<128, 256, 0, stream>>>(h, Wb1, bb1, Apre, NN, DD, DD, DD, DD,
                                       DD, 16, 0);
  gemm16_wmma<<<128, 256, 0, stream>>>(h, Wb1 + (size_t)DD * DD, nullptr, Bpre,
                                       NN, DD, DD, DD, DD, DD, 16, 0);

  // valency path
  valency_hidden_wmma<<<128, 256, 0, stream>>>(h, probs, Wv1, bv1, V1);
  valency_out<<<4, 256, 0, stream>>>(V1, Wv2, bv2, valency);

  // bond pairs: 32736 tiles of 16 pairs, 8 waves/block -> 4092 blocks exact
  bond_wmma<<<4092, 256, 0, stream>>>(Apre, Bpre, Wb2, bb2, bond_logits, pairs);
}